// UnitaryTransformation_91233695301953
// MI455X (gfx1250) — compile-verified
//
#include <hip/hip_runtime.h>
#include <hip/hip_bf16.h>

typedef __attribute__((ext_vector_type(2))) float v2f;
typedef __attribute__((ext_vector_type(4))) float v4f;
typedef __attribute__((ext_vector_type(8))) float v8f;

#define BATCH 32768
#define DIM   1024
#define NUM_GATES 256   // DIM/4
#define NUM_LAYERS 4
#define COL_TILES 64    // DIM/16 : 4 gates per tile
#define ROW_TILES (BATCH/16)

// ---------------------------------------------------------------------------
// Kernel 1: compose the 4 layers of complex 2x2 unitaries into one real 4x4
// matrix per gate, and scatter it into the lane-packed WMMA A-operand layout:
//   Apk[tile t][chunk j][lane L][v]  (64 tiles * 4 chunks * 32 lanes * 2 floats)
// A chunk j is the 16x4 K-slab of the 16x16 block-diagonal matrix; lane L
// holds row m = L&15, k = 2*(L>>4)+v  (per CDNA5 ISA 16x4 f32 A layout).
// ---------------------------------------------------------------------------
__global__ void build_gate_mats(const float* __restrict__ W, float* __restrict__ Apk) {
    int g = blockIdx.x * blockDim.x + threadIdx.x;
    if (g >= NUM_GATES) return;

    // running complex 2x2 product, start = identity
    float u11r = 1.f, u11i = 0.f, u12r = 0.f, u12i = 0.f;
    float u21r = 0.f, u21i = 0.f, u22r = 1.f, u22i = 0.f;

    for (int l = 0; l < NUM_LAYERS; ++l) {
        const float* p = W + ((size_t)l * NUM_GATES + g) * 6;
        float a = p[0], b = p[1], c = p[2], d = p[3], e = p[4];
        float cb = cosf(b), sb = sinf(b);
        float v11r = cosf(a) * cb, v11i = sinf(a) * cb;
        float v12r = cosf(c) * sb, v12i = sinf(c) * sb;
        float v21r = cosf(d) * sb, v21i = sinf(d) * sb;
        float v22r = cosf(e) * cb, v22i = sinf(e) * cb;
        // new = V * U  (layer applied after accumulated product)
        float n11r = v11r*u11r - v11i*u11i + v12r*u21r - v12i*u21i;
        float n11i = v11r*u11i + v11i*u11r + v12r*u21i + v12i*u21r;
        float n12r = v11r*u12r - v11i*u12i + v12r*u22r - v12i*u22i;
        float n12i = v11r*u12i + v11i*u12r + v12r*u22i + v12i*u22r;
        float n21r = v21r*u11r - v21i*u11i + v22r*u21r - v22i*u21i;
        float n21i = v21r*u11i + v21i*u11r + v22r*u21i + v22i*u21r;
        float n22r = v21r*u12r - v21i*u12i + v22r*u22r - v22i*u22i;
        float n22i = v21r*u12i + v21i*u12r + v22r*u22i + v22i*u22r;
        u11r = n11r; u11i = n11i; u12r = n12r; u12i = n12i;
        u21r = n21r; u21i = n21i; u22r = n22r; u22i = n22i;
    }

    // real 4x4 acting on [Re xi, Im xi, Re xj, Im xj]
    float M[4][4] = {
        { u11r, -u11i, u12r, -u12i },
        { u11i,  u11r, u12i,  u12r },
        { u21r, -u21i, u22r, -u22i },
        { u21i,  u21r, u22i,  u22r },
    };

    int t = g >> 2;      // column tile (4 gates per 16-col tile)
    int j = g & 3;       // chunk index (K-slab) within the tile
    float* base = Apk + ((size_t)t * 4 + j) * 64;   // 32 lanes * 2 floats
    for (int L = 0; L < 32; ++L) {
        int m  = L & 15;
        int kh = (L >> 4) * 2;
        float v0 = 0.f, v1 = 0.f;
        if ((m >> 2) == j) { v0 = M[m & 3][kh]; v1 = M[m & 3][kh + 1]; }
        base[L * 2 + 0] = v0;
        base[L * 2 + 1] = v1;
    }
}

// ---------------------------------------------------------------------------
// Kernel 2: streaming transform. One wave per 16x16 tile:
//   D(16x16) = sum_{j=0..3} A_j(16x4) * B_j(4x16)  via v_wmma_f32_16x16x4_f32
// A_j: preloaded lane-packed slabs of the block-diagonal gate matrix.
// B_j: x data, lane L loads 8B at x[row L&15, c0 + 4j + 2*(L>>4)].
// D:   lane n holds out[row n, cols base+8h .. base+8h+7] -> two b128 stores.
// ---------------------------------------------------------------------------
__global__ void __launch_bounds__(256)
apply_gates_wmma(const float* __restrict__ x, const float* __restrict__ Apk,
                 float* __restrict__ out) {
    int wid  = (blockIdx.x * blockDim.x + threadIdx.x) >> 5;
    int lane = threadIdx.x & 31;
    int t = wid & (COL_TILES - 1);   // column tile, fastest varying
    int r = wid >> 6;                // row tile
    int hf = lane >> 4;
    int m  = lane & 15;

    const float* abase = Apk + (size_t)t * 4 * 64;
    v2f a0 = *(const v2f*)(abase + (0 * 32 + lane) * 2);
    v2f a1 = *(const v2f*)(abase + (1 * 32 + lane) * 2);
    v2f a2 = *(const v2f*)(abase + (2 * 32 + lane) * 2);
    v2f a3 = *(const v2f*)(abase + (3 * 32 + lane) * 2);

    const float* xrow = x + ((size_t)r * 16 + m) * DIM + t * 16 + 2 * hf;
    v2f b0 = *(const v2f*)(xrow + 0);
    v2f b1 = *(const v2f*)(xrow + 4);
    v2f b2 = *(const v2f*)(xrow + 8);
    v2f b3 = *(const v2f*)(xrow + 12);

    v8f c = {};
    c = __builtin_amdgcn_wmma_f32_16x16x4_f32(false, a0, false, b0, (short)0, c, false, false);
    c = __builtin_amdgcn_wmma_f32_16x16x4_f32(false, a1, false, b1, (short)0, c, false, false);
    c = __builtin_amdgcn_wmma_f32_16x16x4_f32(false, a2, false, b2, (short)0, c, false, false);
    c = __builtin_amdgcn_wmma_f32_16x16x4_f32(false, a3, false, b3, (short)0, c, false, false);

    float* orow = out + ((size_t)r * 16 + m) * DIM + t * 16 + 8 * hf;
    *(v4f*)(orow + 0) = (v4f){c[0], c[1], c[2], c[3]};
    *(v4f*)(orow + 4) = (v4f){c[4], c[5], c[6], c[7]};
}

extern "C" void kernel_launch(void* const* d_in, const int* in_sizes, int n_in,
                              void* d_out, int out_size, void* d_ws, size_t ws_size,
                              hipStream_t stream) {
    const float* x = (const float*)d_in[0];
    const float* W = (const float*)d_in[1];
    float* out = (float*)d_out;
    float* Apk = (float*)d_ws;   // 64 tiles * 4 chunks * 32 lanes * 2 floats = 64 KB

    build_gate_mats<<<1, NUM_GATES, 0, stream>>>(W, Apk);

    int total_waves = ROW_TILES * COL_TILES;       // 131072
    int blocks = total_waves / 8;                  // 8 waves (256 thr) per block
    apply_gates_wmma<<<blocks, 256, 0, stream>>>(x, Apk, out);
}